// ViolationInformedLossAccelerated_35631048687983
// MI455X (gfx1250) — compile-verified
//
#include <hip/hip_runtime.h>
#include <math.h>

typedef __attribute__((ext_vector_type(2))) float v2f;
typedef __attribute__((ext_vector_type(8))) float v8f;

#define BATCH   2048
#define SN      20
#define KN      128
#define NN      256
#define MTOP    10
#define LSTRIDE 132   // 132 % 64 == 4 -> conflict-free LDS rows

__global__ void vil_init(float* acc) {
    acc[0] = 0.0f;
    acc[1] = 0.0f;
}

__global__ void __launch_bounds__(256)
vil_main(const float* __restrict__ y_pred, const float* __restrict__ y_true,
         const float* __restrict__ P, const float* __restrict__ samples,
         float* __restrict__ acc)
{
    __shared__ float Asm[32 * LSTRIDE];      // samples staged (rows 20..31 zero)
    __shared__ float Cmag[SN * LSTRIDE];     // |codeword| for the P-columns
    __shared__ float wmaxsh[8];

    const int b    = blockIdx.x;
    const int tid  = threadIdx.x;
    const int lane = tid & 31;
    const int w    = tid >> 5;               // wave id, 8 waves

    // ---- stage samples -> LDS (zero-pad rows 20..31 for the M=16..31 tile) ----
    const float* sb = samples + (size_t)b * SN * KN;
    for (int i = tid; i < SN * KN; i += 256) {
        int s = i >> 7, k = i & 127;
        Asm[s * LSTRIDE + k] = sb[i];
    }
    for (int i = tid; i < (32 - SN) * KN; i += 256) {
        int s = SN + (i >> 7), k = i & 127;
        Asm[s * LSTRIDE + k] = 0.0f;
    }
    __syncthreads();

    // ---- GEMM: wave w computes columns [16w,16w+16) of samples(32x128) @ P_b(128x128) ----
    const float* Pb    = P + (size_t)b * KN * (NN - KN);
    const int    col0  = w * 16;
    const int    hlane = lane & 15;
    const int    koff  = (lane >> 4) << 1;   // lanes 0-15: K+0/K+1, lanes 16-31: K+2/K+3

    v8f c0 = {};
    v8f c1 = {};
    #pragma unroll 4
    for (int t = 0; t < 32; ++t) {
        const int kb = 4 * t + koff;
        v2f bf;
        bf.x = Pb[(size_t)kb * 128 + col0 + hlane];
        bf.y = Pb[(size_t)(kb + 1) * 128 + col0 + hlane];
        const float* ar0 = &Asm[hlane * LSTRIDE + kb];
        const float* ar1 = &Asm[(hlane + 16) * LSTRIDE + kb];
        v2f a0; a0.x = ar0[0]; a0.y = ar0[1];
        v2f a1; a1.x = ar1[0]; a1.y = ar1[1];
        c0 = __builtin_amdgcn_wmma_f32_16x16x4_f32(false, a0, false, bf, (short)0, c0, false, false);
        c1 = __builtin_amdgcn_wmma_f32_16x16x4_f32(false, a1, false, bf, (short)0, c1, false, false);
    }

    // ---- store |D| for valid rows (s < 20) ----
    const int rbase = (lane >> 4) << 3;      // C/D layout: lanes 16-31 hold M = r+8
    #pragma unroll
    for (int r = 0; r < 8; ++r) {
        int row = r + rbase;
        Cmag[row * LSTRIDE + col0 + hlane] = fabsf(c0[r]);
        int row1 = row + 16;
        if (row1 < SN)
            Cmag[row1 * LSTRIDE + col0 + hlane] = fabsf(c1[r]);
    }
    __syncthreads();

    // ---- wave-cooperative top-11 per sample row (256 mags: 128 |samples| + 128 |S@P|) ----
    float rowmax = -1.0e30f;
    for (int s = w; s < SN; s += 8) {
        float v[8];
        #pragma unroll
        for (int j = 0; j < 4; ++j) v[j]     = fabsf(Asm[s * LSTRIDE + 4 * lane + j]);
        #pragma unroll
        for (int j = 0; j < 4; ++j) v[4 + j] = Cmag[s * LSTRIDE + 4 * lane + j];

        float top0 = 0.0f, top10 = 0.0f;
        for (int it = 0; it <= MTOP; ++it) {
            float lmax = v[0];
            #pragma unroll
            for (int j = 1; j < 8; ++j) lmax = fmaxf(lmax, v[j]);
            float gmax = lmax;
            #pragma unroll
            for (int off = 16; off > 0; off >>= 1)
                gmax = fmaxf(gmax, __shfl_xor(gmax, off, 32));
            int cand = (lmax == gmax) ? lane : 64;
            #pragma unroll
            for (int off = 16; off > 0; off >>= 1)
                cand = min(cand, __shfl_xor(cand, off, 32));
            if (lane == cand) {               // remove exactly one instance (static indices)
                bool done = false;
                #pragma unroll
                for (int j = 0; j < 8; ++j) {
                    if (!done && v[j] == gmax) { v[j] = -1.0f; done = true; }
                }
            }
            if (it == 0)    top0  = gmax;
            if (it == MTOP) top10 = gmax;
        }
        float hm = top0 / (top10 + 1e-9f);
        rowmax = fmaxf(rowmax, hm);
    }
    if (lane == 0) wmaxsh[w] = rowmax;
    __syncthreads();

    // ---- per-batch scalar tail ----
    if (tid == 0) {
        float mh = wmaxsh[0];
        #pragma unroll
        for (int i = 1; i < 8; ++i) mh = fmaxf(mh, wmaxsh[i]);
        float pen = fmaxf(mh - y_pred[b], 0.0f);
        float yp  = fmaxf(y_pred[b], 1e-9f);
        float yt  = fmaxf(y_true[b], 1e-9f);
        float d   = log2f(yt) - log2f(yp);
        atomicAdd(&acc[0], d * d);
        atomicAdd(&acc[1], pen);
    }
}

__global__ void vil_final(const float* __restrict__ acc, float* __restrict__ out) {
    float logmse = acc[0] / (float)BATCH;
    float viol   = acc[1] / (float)BATCH;
    out[0] = logmse + 0.5f * viol;   // total_loss
    out[1] = logmse;
    out[2] = viol;
}

extern "C" void kernel_launch(void* const* d_in, const int* in_sizes, int n_in,
                              void* d_out, int out_size, void* d_ws, size_t ws_size,
                              hipStream_t stream) {
    const float* y_pred  = (const float*)d_in[0];
    const float* y_true  = (const float*)d_in[1];
    const float* P       = (const float*)d_in[2];
    const float* samples = (const float*)d_in[3];
    // d_in[4] is m (==10), baked in as MTOP
    float* acc = (float*)d_ws;

    vil_init<<<1, 1, 0, stream>>>(acc);
    vil_main<<<BATCH, 256, 0, stream>>>(y_pred, y_true, P, samples, acc);
    vil_final<<<1, 1, 0, stream>>>(acc, (float*)d_out);
}